// ActorMPC_14413910246015
// MI455X (gfx1250) — compile-verified
//
#include <hip/hip_runtime.h>
#include <math.h>

// ---------------------------------------------------------------------------
// ActorMPC on MI455X (gfx1250, wave32).  All matmuls via V_WMMA_F32_16X16X4_F32
// (fp32 kept end-to-end: the Riccati recursion + Cholesky is precision
// sensitive).  GEMM B-panels are staged into LDS with the CDNA5 async copy
// path (global_load_async_to_lds_b128 / ASYNCcnt), shared by 4 waves per
// block.  Workload is compute-bound: ~30 GFLOP fp32 vs ~150MB of traffic
// (everything L2-resident at 192MB).
// ---------------------------------------------------------------------------

typedef __attribute__((ext_vector_type(2))) float v2f;
typedef __attribute__((ext_vector_type(8))) float v8f;

#define BATCH 2048
#define NXc   32
#define NUc   8
#define Tc    50
#define NTAU  40      // nx+nu
#define NPAD  48      // padded to 3x16 tiles (zero-filled) so WMMA runs EXEC=all-1s
#define DTc   0.05f
#define REGc  0.01f
#define LOG2PIc 1.8378770664093453f

__device__ __forceinline__ v8f wmma4(v2f a, v2f b, v8f c) {
  // D = A(16x4,f32) * B(4x16,f32) + C(16x16,f32)
  return __builtin_amdgcn_wmma_f32_16x16x4_f32(
      false, a, false, b, (short)0, c, false, false);
}

__device__ __forceinline__ float softplusf(float x) {
  return fmaxf(x, 0.f) + log1pf(expf(-fabsf(x)));
}

// ---------------------------------------------------------------------------
// GEMM: C[M,N] = epilogue(A[M,K] @ W[K,N] + bias[N])
// Block = 128 threads (4 waves) -> 64x64 C tile, each wave owns a 16x64 strip.
// Per 32-deep k-chunk the B panel (32x64) is async-copied to LDS
// (global_load_async_to_lds_b128, ASYNCcnt) and shared by all 4 waves; A is
// read directly (8B/lane/k-step, reused across 4 n-tiles, L1/L2 friendly).
// EPI==0: ReLU.   EPI==1: col<2000 -> softplus(.)+0.01 (q), else raw (p).
// ---------------------------------------------------------------------------
template <int EPI>
__global__ __launch_bounds__(128) void gemm_wmma(
    const float* __restrict__ Amat, const float* __restrict__ W,
    const float* __restrict__ bias, float* __restrict__ Cmat,
    int M, int N, int K) {
  __shared__ float sB[32][64];  // 8KB B panel

  const int tid  = threadIdx.x;
  const int lane = tid & 31;
  const int w    = tid >> 5;        // wave id 0..3 (uniform per wave)
  const int r    = lane & 15;
  const int kh   = lane >> 4;
  const int m0   = blockIdx.x * 64 + w * 16;
  const int n0   = blockIdx.y * 64;

  // tile guards: uniform across each wave (blockIdx-derived only)
  const bool t0 = (n0 +  0) < N;
  const bool t1 = (n0 + 16) < N;
  const bool t2 = (n0 + 32) < N;
  const bool t3 = (n0 + 48) < N;

  v8f acc0 = {}; v8f acc1 = {}; v8f acc2 = {}; v8f acc3 = {};
  const float* __restrict__ arow = Amat + (size_t)(m0 + r) * K;

  for (int k0 = 0; k0 < K; k0 += 32) {
    // ---- async-stage B[k0:k0+32, n0:n0+64] into LDS (4x b128 per thread) ----
#pragma unroll
    for (int it = 0; it < 4; ++it) {
      const int idx = tid + 128 * it;   // 0..511 b128 transfers
      const int row = idx >> 4;         // 0..31
      const int col = (idx & 15) * 4;   // 0..60, 16B aligned
      if (n0 + col < N) {               // N is 16-aligned: OOB cols only hit skipped tiles
        unsigned ldsa = (unsigned)(uintptr_t)&sB[row][col];
        const float* gp = W + (size_t)(k0 + row) * N + (n0 + col);
        asm volatile("global_load_async_to_lds_b128 %0, %1, off"
                     :: "v"(ldsa), "v"(gp) : "memory");
      }
    }
    if (k0 + 32 < K) __builtin_prefetch(arow + k0 + 32, 0, 0);  // speculative: fault-safe
    asm volatile("s_wait_asynccnt 0x0" ::: "memory");
    __syncthreads();

    // ---- 8 k-steps of WMMA out of LDS ----
#pragma unroll
    for (int kk = 0; kk < 32; kk += 4) {
      v2f a;
      a.x = arow[k0 + kk + 2 * kh + 0];
      a.y = arow[k0 + kk + 2 * kh + 1];
      if (t0) { v2f b; b.x = sB[kk + 2*kh][ 0 + r]; b.y = sB[kk + 2*kh + 1][ 0 + r]; acc0 = wmma4(a, b, acc0); }
      if (t1) { v2f b; b.x = sB[kk + 2*kh][16 + r]; b.y = sB[kk + 2*kh + 1][16 + r]; acc1 = wmma4(a, b, acc1); }
      if (t2) { v2f b; b.x = sB[kk + 2*kh][32 + r]; b.y = sB[kk + 2*kh + 1][32 + r]; acc2 = wmma4(a, b, acc2); }
      if (t3) { v2f b; b.x = sB[kk + 2*kh][48 + r]; b.y = sB[kk + 2*kh + 1][48 + r]; acc3 = wmma4(a, b, acc3); }
    }
    __syncthreads();
  }

  // ---- epilogue: bias + activation, direct store ----
  const int mb = m0 + 8 * kh;
  v8f* accs[4] = {&acc0, &acc1, &acc2, &acc3};
  const bool tg[4] = {t0, t1, t2, t3};
#pragma unroll
  for (int i = 0; i < 4; ++i) {
    if (!tg[i]) continue;
    const int n = n0 + 16 * i + r;
    const float bv = bias[n];
    v8f acc = *accs[i];
#pragma unroll
    for (int j = 0; j < 8; ++j) {
      float v = acc[j] + bv;
      if (EPI == 0) {
        v = fmaxf(v, 0.f);
      } else {
        if (n < 2000) v = softplusf(v) + 0.01f;  // q region; p region raw
      }
      Cmat[(size_t)(mb + j) * N + n] = v;
    }
  }
}

// ---------------------------------------------------------------------------
// LQR kernel: one block (64 threads = 2 waves) per batch element.
// Backward Riccati: per step the three matmuls run as WMMA f32 16x16x4 tiles
// out of LDS (dims padded to 48 with zeros), Quu solved by Cholesky.
// K_t/k_t spilled to workspace; forward rollout reads them back in time order.
// ---------------------------------------------------------------------------
__global__ __launch_bounds__(64) void lqr_kernel(
    const float* __restrict__ x0g, const float* __restrict__ qp,
    const float* __restrict__ Ag,  const float* __restrict__ Bmg,
    const float* __restrict__ log_std, const float* __restrict__ noise,
    float* __restrict__ Kbuf, float* __restrict__ kbuf,
    float* __restrict__ out) {
  __shared__ float sF[NXc][NPAD];    // F = [I+dt*A, dt*B], zero-padded cols 40..47
  __shared__ float sV[NXc][NXc];
  __shared__ float sG[NXc][NPAD];    // G = V*F
  __shared__ float sQ[NPAD][NPAD];   // Q = F^T*G + diag(qt)
  __shared__ float sK[NUc][NXc];
  __shared__ float sVn[NXc][NXc];
  __shared__ float sLc[NUc][NUc];
  __shared__ float sqv[NTAU];
  __shared__ float skt[NUc];
  __shared__ float stau[NTAU];
  __shared__ float sv[NXc];
  __shared__ float sxn[NXc];

  const int tid  = threadIdx.x;
  const int lane = tid & 31;
  const int w    = tid >> 5;       // wave id 0/1 (uniform per wave)
  const int r    = lane & 15;
  const int kh   = lane >> 4;
  const int b    = blockIdx.x;

  // Build padded F.
  for (int e = tid; e < NXc * NPAD; e += 64) {
    int i = e / NPAD, j = e % NPAD;
    float f = 0.f;
    if (j < NXc)       f = ((i == j) ? 1.f : 0.f) + DTc * Ag[i * NXc + j];
    else if (j < NTAU) f = DTc * Bmg[i * NUc + (j - NXc)];
    sF[i][j] = f;
  }
  const float* __restrict__ qpb = qp + (size_t)b * 4000;
  // V0 = diag(q[T-1,:nx]); v0 = p[T-1,:nx]
  for (int e = tid; e < NXc * NXc; e += 64) {
    int i = e >> 5, j = e & 31;
    sV[i][j] = (i == j) ? qpb[(Tc - 1) * NTAU + i] : 0.f;
  }
  if (tid < NXc) sv[tid] = qpb[2000 + (Tc - 1) * NTAU + tid];
  __syncthreads();

  // ---------------- backward pass (t = T-1 .. 0) ----------------
  for (int t = Tc - 1; t >= 0; --t) {
    const float* __restrict__ qt = qpb + t * NTAU;
    const float* __restrict__ pt = qpb + 2000 + t * NTAU;

    // G = V * F   (32x32 @ 32x48): 2x3 tiles split across the 2 waves.
    for (int tile = w; tile < 6; tile += 2) {
      const int m0 = (tile / 3) * 16, n0 = (tile % 3) * 16;
      v8f acc = {};
      for (int k = 0; k < NXc; k += 4) {
        v2f a; a.x = sV[m0 + r][k + 2 * kh];     a.y = sV[m0 + r][k + 2 * kh + 1];
        v2f bb; bb.x = sF[k + 2 * kh][n0 + r];   bb.y = sF[k + 2 * kh + 1][n0 + r];
        acc = wmma4(a, bb, acc);
      }
#pragma unroll
      for (int j = 0; j < 8; ++j) sG[m0 + 8 * kh + j][n0 + r] = acc[j];
    }
    // qv[j] = pt[j] + sum_x F[x][j] * v[x]
    if (tid < NTAU) {
      float s = pt[tid];
      for (int xx = 0; xx < NXc; ++xx) s += sF[xx][tid] * sv[xx];
      sqv[tid] = s;
    }
    __syncthreads();

    // Q = F^T * G  (48x32 @ 32x48): 3x3 tiles.
    for (int tile = w; tile < 9; tile += 2) {
      const int m0 = (tile / 3) * 16, n0 = (tile % 3) * 16;
      v8f acc = {};
      for (int k = 0; k < NXc; k += 4) {
        v2f a; a.x = sF[k + 2 * kh][m0 + r];     a.y = sF[k + 2 * kh + 1][m0 + r];
        v2f bb; bb.x = sG[k + 2 * kh][n0 + r];   bb.y = sG[k + 2 * kh + 1][n0 + r];
        acc = wmma4(a, bb, acc);
      }
#pragma unroll
      for (int j = 0; j < 8; ++j) sQ[m0 + 8 * kh + j][n0 + r] = acc[j];
    }
    __syncthreads();
    if (tid < NTAU) sQ[tid][tid] += qt[tid];
    __syncthreads();

    // Cholesky of Quu = Q[32:40,32:40] + REG*I  (tiny: one thread)
    if (tid == 0) {
      float Lc[NUc][NUc];
      for (int i = 0; i < NUc; ++i)
        for (int j = 0; j <= i; ++j) {
          float s = sQ[NXc + i][NXc + j] + ((i == j) ? REGc : 0.f);
          for (int m = 0; m < j; ++m) s -= Lc[i][m] * Lc[j][m];
          Lc[i][j] = (i == j) ? sqrtf(s) : s / Lc[j][j];
        }
      for (int i = 0; i < NUc; ++i)
        for (int j = 0; j <= i; ++j) sLc[i][j] = Lc[i][j];
    }
    __syncthreads();

    // 33 triangular solves in parallel: K = -Quu^{-1} Qux, k = -Quu^{-1} qv_u.
    if (tid < NXc + 1) {
      float rhs[NUc], y[NUc], z[NUc];
#pragma unroll
      for (int u = 0; u < NUc; ++u)
        rhs[u] = (tid < NXc) ? sQ[NXc + u][tid] : sqv[NXc + u];
#pragma unroll
      for (int i = 0; i < NUc; ++i) {
        float s = rhs[i];
        for (int m = 0; m < i; ++m) s -= sLc[i][m] * y[m];
        y[i] = s / sLc[i][i];
      }
#pragma unroll
      for (int i = NUc - 1; i >= 0; --i) {
        float s = y[i];
        for (int m = i + 1; m < NUc; ++m) s -= sLc[m][i] * z[m];
        z[i] = s / sLc[i][i];
      }
      if (tid < NXc) {
#pragma unroll
        for (int u = 0; u < NUc; ++u) sK[u][tid] = -z[u];
      } else {
#pragma unroll
        for (int u = 0; u < NUc; ++u) skt[u] = -z[u];
      }
    }
    __syncthreads();

    // Spill gains for the forward pass.
    {
      const size_t base = (size_t)b * Tc + t;
      for (int e = tid; e < NUc * NXc; e += 64)
        Kbuf[base * 256 + e] = sK[e >> 5][e & 31];
      if (tid < NUc) kbuf[base * NUc + tid] = skt[tid];
    }

    // Vn = Qxx + Qxu*K  (32x32 += 32x8 @ 8x32): 2x2 tiles, C preloaded = Qxx.
    for (int tile = w; tile < 4; tile += 2) {
      const int m0 = (tile >> 1) * 16, n0 = (tile & 1) * 16;
      v8f acc;
#pragma unroll
      for (int j = 0; j < 8; ++j) acc[j] = sQ[m0 + 8 * kh + j][n0 + r];
      for (int k = 0; k < NUc; k += 4) {
        // A = Qxu: A[m][k] = Q[32+k][m]
        v2f a; a.x = sQ[NXc + k + 2 * kh][m0 + r];  a.y = sQ[NXc + k + 2 * kh + 1][m0 + r];
        v2f bb; bb.x = sK[k + 2 * kh][n0 + r];      bb.y = sK[k + 2 * kh + 1][n0 + r];
        acc = wmma4(a, bb, acc);
      }
#pragma unroll
      for (int j = 0; j < 8; ++j) sVn[m0 + 8 * kh + j][n0 + r] = acc[j];
    }
    __syncthreads();

    // V = 0.5*(Vn + Vn^T);  v = qv_x + Qxu*k
    for (int e = tid; e < NXc * NXc; e += 64) {
      int i = e >> 5, j = e & 31;
      sV[i][j] = 0.5f * (sVn[i][j] + sVn[j][i]);
    }
    if (tid < NXc) {
      float s = sqv[tid];
#pragma unroll
      for (int u = 0; u < NUc; ++u) s += sQ[NXc + u][tid] * skt[u];
      sv[tid] = s;
    }
    __syncthreads();
  }

  // ---------------- forward rollout ----------------
  float* __restrict__ actionO = out;                              // (B,8)
  float* __restrict__ lpO     = out + (size_t)BATCH * NUc;        // (B,)
  float* __restrict__ XO      = lpO + BATCH;                      // (B,51,32)
  float* __restrict__ UO      = XO + (size_t)BATCH * (Tc + 1) * NXc;  // (B,50,8)

  const float* __restrict__ xb = x0g + (size_t)b * NXc;
  if (tid < NXc) {
    stau[tid] = xb[tid];
    XO[((size_t)b * (Tc + 1)) * NXc + tid] = xb[tid];
  }
  __syncthreads();

  for (int t = 0; t < Tc; ++t) {
    const size_t base = (size_t)b * Tc + t;
    if (tid < NUc) {
      float s = kbuf[base * NUc + tid];
      const float* __restrict__ Kr = Kbuf + base * 256 + tid * NXc;
      for (int xx = 0; xx < NXc; ++xx) s += Kr[xx] * stau[xx];
      stau[NXc + tid] = s;
      UO[base * NUc + tid] = s;
    }
    __syncthreads();
    if (t == 0) {
      if (tid < NUc)
        actionO[(size_t)b * NUc + tid] =
            stau[NXc + tid] + expf(log_std[tid]) * noise[(size_t)b * NUc + tid];
      if (tid == 0) {
        float lp = 0.f;
        for (int u = 0; u < NUc; ++u) {
          float nz = noise[(size_t)b * NUc + u];
          lp += -0.5f * nz * nz - log_std[u] - 0.5f * LOG2PIc;
        }
        lpO[b] = lp;
      }
    }
    if (tid < NXc) {
      float s = 0.f;
      for (int j = 0; j < NTAU; ++j) s += sF[tid][j] * stau[j];
      sxn[tid] = s;
      XO[((size_t)b * (Tc + 1) + t + 1) * NXc + tid] = s;
    }
    __syncthreads();
    if (tid < NXc) stau[tid] = sxn[tid];
    __syncthreads();
  }
}

// ---------------------------------------------------------------------------
extern "C" void kernel_launch(void* const* d_in, const int* in_sizes, int n_in,
                              void* d_out, int out_size, void* d_ws, size_t ws_size,
                              hipStream_t stream) {
  const float* x       = (const float*)d_in[0];
  const float* W1      = (const float*)d_in[1];
  const float* b1      = (const float*)d_in[2];
  const float* W2      = (const float*)d_in[3];
  const float* b2      = (const float*)d_in[4];
  const float* W3      = (const float*)d_in[5];
  const float* b3      = (const float*)d_in[6];
  const float* log_std = (const float*)d_in[7];
  const float* A       = (const float*)d_in[8];
  const float* Bm      = (const float*)d_in[9];
  const float* noise   = (const float*)d_in[10];
  float* out = (float*)d_out;

  // Workspace layout (floats): h1 | h2 | qp(q softplus'ed + p) | Kbuf | kbuf
  float* h1 = (float*)d_ws;                       // 2048*512
  float* h2 = h1 + (size_t)BATCH * 512;           // 2048*512
  float* qp = h2 + (size_t)BATCH * 512;           // 2048*4000
  float* Kb = qp + (size_t)BATCH * 4000;          // 2048*50*256
  float* kb = Kb + (size_t)BATCH * Tc * 256;      // 2048*50*8

  // MLP: fp32 WMMA GEMMs, async LDS staging, fused bias+activation epilogues.
  gemm_wmma<0><<<dim3(BATCH / 64, 512 / 64), 128, 0, stream>>>(x,  W1, b1, h1, BATCH, 512, 32);
  gemm_wmma<0><<<dim3(BATCH / 64, 512 / 64), 128, 0, stream>>>(h1, W2, b2, h2, BATCH, 512, 512);
  gemm_wmma<1><<<dim3(BATCH / 64, (4000 + 63) / 64), 128, 0, stream>>>(h2, W3, b3, qp, BATCH, 4000, 512);

  // LQR backward+forward, one workgroup per batch element.
  lqr_kernel<<<BATCH, 64, 0, stream>>>(x, qp, A, Bm, log_std, noise, Kb, kb, out);
}